// SphericalConv_70961449664970
// MI455X (gfx1250) — compile-verified
//
#include <hip/hip_runtime.h>

typedef __attribute__((ext_vector_type(2))) float v2f;
typedef __attribute__((ext_vector_type(4))) float v4f;
typedef __attribute__((ext_vector_type(8))) float v8f;

// D = A(16x4,f32) * B(4x16,f32) + C(16x16,f32)  -- CDNA5 V_WMMA_F32_16X16X4_F32
__device__ __forceinline__ v8f wmma4(v2f a, v2f b, v8f c) {
  return __builtin_amdgcn_wmma_f32_16x16x4_f32(false, a, false, b, (short)0, c, false, false);
}

__device__ __forceinline__ float silu_c(float x) {
  return (x / (1.0f + __expf(-x))) * 1.679177f;
}

__global__ void zero_kernel(v4f* __restrict__ p, long n4) {
  long i = (long)blockIdx.x * blockDim.x + threadIdx.x;
  long stride = (long)gridDim.x * blockDim.x;
  const v4f z = {};
  for (; i < n4; i += stride) p[i] = z;
}

// ---------------------------------------------------------------------------
// K1a: v0 = x0 @ W_val0 / sqrt(128)        (contiguous A)
// ---------------------------------------------------------------------------
__global__ __launch_bounds__(32)
void nodeval0_kernel(const float* __restrict__ nf, const float* __restrict__ Wv0,
                     float* __restrict__ v0ws) {
  const int lane = threadIdx.x;
  const bool lo = lane < 16;
  const int nsel = lane & 15;
  const int koff = lo ? 0 : 2;
  const int n0 = blockIdx.x * 16;
  const float* aptr = nf + (long)(n0 + nsel) * 512;

  v8f acc[8] = {};
#pragma unroll 1
  for (int k = 0; k < 32; ++k) {
    const int kr = k * 4 + koff;
    v2f a;
    a.x = aptr[kr];
    a.y = aptr[kr + 1];
#pragma unroll
    for (int t = 0; t < 8; ++t) {
      const int col = t * 16 + nsel;
      v2f b;
      b.x = Wv0[kr * 128 + col];
      b.y = Wv0[(kr + 1) * 128 + col];
      acc[t] = wmma4(a, b, acc[t]);
    }
  }
  const float s = 0.08838834764831845f; // 1/sqrt(128)
#pragma unroll
  for (int t = 0; t < 8; ++t)
#pragma unroll
    for (int r = 0; r < 8; ++r) {
      const int row = n0 + r + (lo ? 0 : 8);
      v0ws[(long)row * 128 + t * 16 + nsel] = acc[t][r] * s;
    }
}

// ---------------------------------------------------------------------------
// K1b: v1[n][m][w] = sum_u x1[n,u,m] W_val1[u,w] / sqrt(128)   (A stride 3)
// grid (N/16, 3)
// ---------------------------------------------------------------------------
__global__ __launch_bounds__(32)
void nodeval1_kernel(const float* __restrict__ nf, const float* __restrict__ Wv1,
                     float* __restrict__ v1ws) {
  const int lane = threadIdx.x;
  const bool lo = lane < 16;
  const int nsel = lane & 15;
  const int koff = lo ? 0 : 2;
  const int n0 = blockIdx.x * 16;
  const int m = blockIdx.y;
  const float* aptr = nf + (long)(n0 + nsel) * 512 + 128 + m;

  v8f acc[8] = {};
#pragma unroll 1
  for (int k = 0; k < 32; ++k) {
    const int kr = k * 4 + koff;
    v2f a;
    a.x = aptr[kr * 3];
    a.y = aptr[kr * 3 + 3];
#pragma unroll
    for (int t = 0; t < 8; ++t) {
      const int col = t * 16 + nsel;
      v2f b;
      b.x = Wv1[kr * 128 + col];
      b.y = Wv1[(kr + 1) * 128 + col];
      acc[t] = wmma4(a, b, acc[t]);
    }
  }
  const float s = 0.08838834764831845f;
#pragma unroll
  for (int t = 0; t < 8; ++t)
#pragma unroll
    for (int r = 0; r < 8; ++r) {
      const int row = n0 + r + (lo ? 0 : 8);
      v1ws[(long)row * 384 + m * 128 + t * 16 + nsel] = acc[t][r] * s;
    }
}

// ---------------------------------------------------------------------------
// K2: fused radial MLP + message formation + scatter-add (16 edges per wave)
// ---------------------------------------------------------------------------
__global__ __launch_bounds__(32)
void edge_kernel(const float* __restrict__ ef, const float* __restrict__ ea,
                 const int* __restrict__ eidx, int E,
                 const float* __restrict__ Wr0, const float* __restrict__ Wr1,
                 const float* __restrict__ Wr2, const float* __restrict__ Wr3,
                 const float* __restrict__ v0ws, const float* __restrict__ v1ws,
                 float* __restrict__ message) {
  __shared__ float sh_h[16][68];    // MLP hidden (16 edges x 64), +4 pad
  __shared__ float sh_s0[16][132];  // gathered v0[sender]         (row = 528B, 16B aligned)
  __shared__ float sh_sd[16][132];  // sum_m v1[sender][m][u]*ea1[m]
  __shared__ int   sSnd[16];
  __shared__ int   sRcv[16];
  __shared__ float sEa0[16];
  __shared__ float sEa1[16][3];

  const int lane = threadIdx.x;
  const bool lo = lane < 16;
  const int nsel = lane & 15;
  const int koff = lo ? 0 : 2;
  const long e0 = (long)blockIdx.x * 16;

  if (lo) {
    const long e = e0 + lane;
    sSnd[lane] = eidx[e];
    sRcv[lane] = eidx[E + e];
    const v4f eav = *(const v4f*)(ea + e * 4);   // 16B-aligned, one b128
    sEa0[lane] = eav.x;
    sEa1[lane][0] = eav.y;
    sEa1[lane][1] = eav.z;
    sEa1[lane][2] = eav.w;
  }
  __syncthreads();

  // gather s0 and ea1-contracted s1 into LDS: one float4 lane-slice per row
  for (int r = 0; r < 16; ++r) {
    const v4f* v0r = (const v4f*)(v0ws + (long)sSnd[r] * 128);
    const v4f* v1r = (const v4f*)(v1ws + (long)sSnd[r] * 384);
    const float a0 = sEa1[r][0], a1 = sEa1[r][1], a2 = sEa1[r][2];
    const v4f p0 = v0r[lane];
    const v4f q0 = v1r[lane];
    const v4f q1 = v1r[32 + lane];
    const v4f q2 = v1r[64 + lane];
    *(v4f*)&sh_s0[r][lane * 4] = p0;
    v4f d;
    d.x = q0.x * a0 + q1.x * a1 + q2.x * a2;
    d.y = q0.y * a0 + q1.y * a1 + q2.y * a2;
    d.z = q0.z * a0 + q1.z * a1 + q2.z * a2;
    d.w = q0.w * a0 + q1.w * a1 + q2.w * a2;
    *(v4f*)&sh_sd[r][lane * 4] = d;
  }
  __syncthreads();

  // --- MLP layer 0: (16x8) @ (8x64), silu ---
  {
    v8f acc[4] = {};
#pragma unroll
    for (int k = 0; k < 2; ++k) {
      const int kr = k * 4 + koff;
      v2f a;
      a.x = ef[(e0 + nsel) * 8 + kr];
      a.y = ef[(e0 + nsel) * 8 + kr + 1];
#pragma unroll
      for (int t = 0; t < 4; ++t) {
        const int col = t * 16 + nsel;
        v2f b;
        b.x = Wr0[kr * 64 + col];
        b.y = Wr0[(kr + 1) * 64 + col];
        acc[t] = wmma4(a, b, acc[t]);
      }
    }
#pragma unroll
    for (int t = 0; t < 4; ++t)
#pragma unroll
      for (int r = 0; r < 8; ++r) {
        const int row = r + (lo ? 0 : 8);
        sh_h[row][t * 16 + nsel] = silu_c(acc[t][r] * 0.3535533906f); // /sqrt(8)
      }
    __syncthreads();
  }

  // --- MLP layers 1,2: (16x64) @ (64x64), silu, /8 ---
#pragma unroll 1
  for (int layer = 0; layer < 2; ++layer) {
    const float* W = layer ? Wr2 : Wr1;
    v8f acc[4] = {};
#pragma unroll 1
    for (int k = 0; k < 16; ++k) {
      const int kr = k * 4 + koff;
      v2f a;
      a.x = sh_h[nsel][kr];
      a.y = sh_h[nsel][kr + 1];
#pragma unroll
      for (int t = 0; t < 4; ++t) {
        const int col = t * 16 + nsel;
        v2f b;
        b.x = W[kr * 64 + col];
        b.y = W[(kr + 1) * 64 + col];
        acc[t] = wmma4(a, b, acc[t]);
      }
    }
    __syncthreads(); // done reading sh_h
#pragma unroll
    for (int t = 0; t < 4; ++t)
#pragma unroll
      for (int r = 0; r < 8; ++r) {
        const int row = r + (lo ? 0 : 8);
        sh_h[row][t * 16 + nsel] = silu_c(acc[t][r] * 0.125f);
      }
    __syncthreads();
  }

  // --- final layer in 4 column chunks (w_a,w_b,w_c,w_d), fused messages ---
#pragma unroll
  for (int q = 0; q < 4; ++q) {
    v8f acc[8] = {};
#pragma unroll 1
    for (int k = 0; k < 16; ++k) {
      const int kr = k * 4 + koff;
      v2f a;
      a.x = sh_h[nsel][kr];
      a.y = sh_h[nsel][kr + 1];
#pragma unroll
      for (int t = 0; t < 8; ++t) {
        const int col = q * 128 + t * 16 + nsel;
        v2f b;
        b.x = Wr3[kr * 512 + col];
        b.y = Wr3[(kr + 1) * 512 + col];
        acc[t] = wmma4(a, b, acc[t]);
      }
    }
#pragma unroll
    for (int t = 0; t < 8; ++t)
#pragma unroll
      for (int r = 0; r < 8; ++r) {
        const int row = r + (lo ? 0 : 8);
        const int u = t * 16 + nsel;
        const float wv = acc[t][r] * 0.125f; // /8
        float* mrow = message + (long)sRcv[row] * 1024;
        if (q == 0) {
          unsafeAtomicAdd(mrow + u, wv * sh_s0[row][u] * sEa0[row]);
        } else if (q == 1) {
          unsafeAtomicAdd(mrow + 128 + u, wv * sh_sd[row][u] * 0.5773502692f);
        } else if (q == 2) {
          const float t0 = wv * sh_s0[row][u];
          unsafeAtomicAdd(mrow + 256 + u * 3 + 0, t0 * sEa1[row][0]);
          unsafeAtomicAdd(mrow + 256 + u * 3 + 1, t0 * sEa1[row][1]);
          unsafeAtomicAdd(mrow + 256 + u * 3 + 2, t0 * sEa1[row][2]);
        } else {
          const float* v1r = v1ws + (long)sSnd[row] * 384;
          const float we = wv * sEa0[row];
          unsafeAtomicAdd(mrow + 640 + u * 3 + 0, we * v1r[u]);
          unsafeAtomicAdd(mrow + 640 + u * 3 + 1, we * v1r[128 + u]);
          unsafeAtomicAdd(mrow + 640 + u * 3 + 2, we * v1r[256 + u]);
        }
      }
  }
}

// ---------------------------------------------------------------------------
// K3a: msg0 = message[:, :256] @ W_lin0 / 256     (contiguous A)
// ---------------------------------------------------------------------------
__global__ __launch_bounds__(32)
void msglin0_kernel(const float* __restrict__ message, const float* __restrict__ Wl0,
                    float* __restrict__ msg0) {
  const int lane = threadIdx.x;
  const bool lo = lane < 16;
  const int nsel = lane & 15;
  const int koff = lo ? 0 : 2;
  const int n0 = blockIdx.x * 16;
  const float* aptr = message + (long)(n0 + nsel) * 1024;

  v8f acc[8] = {};
#pragma unroll 1
  for (int k = 0; k < 64; ++k) {
    const int kr = k * 4 + koff;
    v2f a;
    a.x = aptr[kr];
    a.y = aptr[kr + 1];
#pragma unroll
    for (int t = 0; t < 8; ++t) {
      const int col = t * 16 + nsel;
      v2f b;
      b.x = Wl0[kr * 128 + col];
      b.y = Wl0[(kr + 1) * 128 + col];
      acc[t] = wmma4(a, b, acc[t]);
    }
  }
  const float s = 1.0f / 256.0f; // 1/sqrt(256)/AVG_NEIGH
#pragma unroll
  for (int t = 0; t < 8; ++t)
#pragma unroll
    for (int r = 0; r < 8; ++r) {
      const int row = n0 + r + (lo ? 0 : 8);
      msg0[(long)row * 128 + t * 16 + nsel] = acc[t][r] * s;
    }
}

// ---------------------------------------------------------------------------
// K3b: msg1[n][m][w] = sum_u message[n,256+u*3+m] W_lin1[u,w] / 256   grid (nT,3)
// ---------------------------------------------------------------------------
__global__ __launch_bounds__(32)
void msglin1_kernel(const float* __restrict__ message, const float* __restrict__ Wl1,
                    float* __restrict__ msg1) {
  const int lane = threadIdx.x;
  const bool lo = lane < 16;
  const int nsel = lane & 15;
  const int koff = lo ? 0 : 2;
  const int n0 = blockIdx.x * 16;
  const int m = blockIdx.y;
  const float* aptr = message + (long)(n0 + nsel) * 1024 + 256 + m;

  v8f acc[8] = {};
#pragma unroll 1
  for (int k = 0; k < 64; ++k) {
    const int kr = k * 4 + koff;
    v2f a;
    a.x = aptr[kr * 3];
    a.y = aptr[kr * 3 + 3];
#pragma unroll
    for (int t = 0; t < 8; ++t) {
      const int col = t * 16 + nsel;
      v2f b;
      b.x = Wl1[kr * 128 + col];
      b.y = Wl1[(kr + 1) * 128 + col];
      acc[t] = wmma4(a, b, acc[t]);
    }
  }
  const float s = 1.0f / 256.0f;
#pragma unroll
  for (int t = 0; t < 8; ++t)
#pragma unroll
    for (int r = 0; r < 8; ++r) {
      const int row = n0 + r + (lo ? 0 : 8);
      msg1[(long)row * 384 + m * 128 + t * 16 + nsel] = acc[t][r] * s;
    }
}

// ---------------------------------------------------------------------------
// K4a: y0 = (msg0 ⊗ attrs) @ W_skip0, K = 1280
// ---------------------------------------------------------------------------
__global__ __launch_bounds__(32)
void skip0_kernel(const float* __restrict__ attrs, const float* __restrict__ msg0,
                  const float* __restrict__ Wsk0, float* __restrict__ out) {
  const int lane = threadIdx.x;
  const bool lo = lane < 16;
  const int nsel = lane & 15;
  const int koff = lo ? 0 : 2;
  const int n0 = blockIdx.x * 16;
  const int rown = n0 + nsel;
  const float* arow = attrs + (long)rown * 10;
  const float* srow = msg0 + (long)rown * 128;

  v8f acc[8] = {};
#pragma unroll 1
  for (int k = 0; k < 320; ++k) {
    const int kr = k * 4 + koff;
    const int u0 = kr / 10, vv0 = kr - u0 * 10;
    const int kr1 = kr + 1;
    const int u1 = kr1 / 10, vv1 = kr1 - u1 * 10;
    v2f a;
    a.x = srow[u0] * arow[vv0];
    a.y = srow[u1] * arow[vv1];
#pragma unroll
    for (int t = 0; t < 8; ++t) {
      const int col = t * 16 + nsel;
      v2f b;
      b.x = Wsk0[kr * 128 + col];
      b.y = Wsk0[kr1 * 128 + col];
      acc[t] = wmma4(a, b, acc[t]);
    }
  }
  const float fn = 0.02795084972f; // 1/sqrt(1280)
#pragma unroll
  for (int t = 0; t < 8; ++t)
#pragma unroll
    for (int r = 0; r < 8; ++r) {
      const int row = n0 + r + (lo ? 0 : 8);
      out[(long)row * 512 + t * 16 + nsel] = acc[t][r] * fn;
    }
}

// ---------------------------------------------------------------------------
// K4b: y1[n][w][m] = (msg1[n][m] ⊗ attrs) @ W_skip1, grid (nT,3)
// ---------------------------------------------------------------------------
__global__ __launch_bounds__(32)
void skip1_kernel(const float* __restrict__ attrs, const float* __restrict__ msg1,
                  const float* __restrict__ Wsk1, float* __restrict__ out) {
  const int lane = threadIdx.x;
  const bool lo = lane < 16;
  const int nsel = lane & 15;
  const int koff = lo ? 0 : 2;
  const int n0 = blockIdx.x * 16;
  const int m = blockIdx.y;
  const int rown = n0 + nsel;
  const float* arow = attrs + (long)rown * 10;
  const float* srow = msg1 + (long)rown * 384 + m * 128;

  v8f acc[8] = {};
#pragma unroll 1
  for (int k = 0; k < 320; ++k) {
    const int kr = k * 4 + koff;
    const int u0 = kr / 10, vv0 = kr - u0 * 10;
    const int kr1 = kr + 1;
    const int u1 = kr1 / 10, vv1 = kr1 - u1 * 10;
    v2f a;
    a.x = srow[u0] * arow[vv0];
    a.y = srow[u1] * arow[vv1];
#pragma unroll
    for (int t = 0; t < 8; ++t) {
      const int col = t * 16 + nsel;
      v2f b;
      b.x = Wsk1[kr * 128 + col];
      b.y = Wsk1[kr1 * 128 + col];
      acc[t] = wmma4(a, b, acc[t]);
    }
  }
  const float fn = 0.02795084972f;
#pragma unroll
  for (int t = 0; t < 8; ++t)
#pragma unroll
    for (int r = 0; r < 8; ++r) {
      const int row = n0 + r + (lo ? 0 : 8);
      out[(long)row * 512 + 128 + (t * 16 + nsel) * 3 + m] = acc[t][r] * fn;
    }
}

extern "C" void kernel_launch(void* const* d_in, const int* in_sizes, int n_in,
                              void* d_out, int out_size, void* d_ws, size_t ws_size,
                              hipStream_t stream) {
  const float* node_attrs = (const float*)d_in[1];
  const float* node_feats = (const float*)d_in[2];
  const float* edge_attrs = (const float*)d_in[3];
  const float* edge_feats = (const float*)d_in[4];
  const int*   edge_index = (const int*)d_in[5];
  const float* W_val0  = (const float*)d_in[6];
  const float* W_val1  = (const float*)d_in[7];
  const float* W_r0    = (const float*)d_in[8];
  const float* W_r1    = (const float*)d_in[9];
  const float* W_r2    = (const float*)d_in[10];
  const float* W_r3    = (const float*)d_in[11];
  const float* W_lin0  = (const float*)d_in[12];
  const float* W_lin1  = (const float*)d_in[13];
  const float* W_skip0 = (const float*)d_in[14];
  const float* W_skip1 = (const float*)d_in[15];
  float* out = (float*)d_out;

  const int N = in_sizes[1] / 10;
  const int E = in_sizes[5] / 2;

  float* ws = (float*)d_ws;
  float* v0ws    = ws;                            // N*128
  float* v1ws    = v0ws + (size_t)N * 128;        // N*384  [n][m][w]
  float* message = v1ws + (size_t)N * 384;        // N*1024
  float* msg0    = message + (size_t)N * 1024;    // N*128
  float* msg1    = msg0 + (size_t)N * 128;        // N*384  [n][m][w]

  const int nT = N / 16;
  const int eT = E / 16;

  zero_kernel<<<2048, 256, 0, stream>>>((v4f*)message, (long)N * 256);
  nodeval0_kernel<<<nT, 32, 0, stream>>>(node_feats, W_val0, v0ws);
  nodeval1_kernel<<<dim3(nT, 3), 32, 0, stream>>>(node_feats, W_val1, v1ws);
  edge_kernel<<<eT, 32, 0, stream>>>(edge_feats, edge_attrs, edge_index, E,
                                     W_r0, W_r1, W_r2, W_r3, v0ws, v1ws, message);
  msglin0_kernel<<<nT, 32, 0, stream>>>(message, W_lin0, msg0);
  msglin1_kernel<<<dim3(nT, 3), 32, 0, stream>>>(message, W_lin1, msg1);
  skip0_kernel<<<nT, 32, 0, stream>>>(node_attrs, msg0, W_skip0, out);
  skip1_kernel<<<dim3(nT, 3), 32, 0, stream>>>(node_attrs, msg1, W_skip1, out);
}